// AdditionLinear_90314572300851
// MI455X (gfx1250) — compile-verified
//
#include <hip/hip_runtime.h>
#include <hip/hip_bf16.h>
#include <stdint.h>

// Problem constants (match reference).
#define N_ROWS   2048   // B*T
#define K_DIM    1024   // IN_F
#define C_COLS   4096   // OUT_F

// Tiling.
#define XT       64     // x rows per block
#define WT       64     // w rows per block
#define KC       32     // K elements per chunk
#define NCHUNK   (K_DIM / KC)          // 32
// TDM inserts pad_amount(=4) dwords after every pad_interval(=32) dwords stored,
// so each 32-float tile row lands at a 36-dword stride in LDS (144 B, 16B-aligned).
#define LDS_STRIDE 36

typedef unsigned int v4u __attribute__((ext_vector_type(4)));
typedef int          v8i __attribute__((ext_vector_type(8)));
typedef int          v4i __attribute__((ext_vector_type(4)));

// Issue one TDM tile load: 2D tensor (rows x K_DIM f32, row stride K_DIM),
// tile = KC x 64 rows, into LDS at lds_byte_off with padded rows.
__device__ __forceinline__ void tdm_load_tile(const float* gbase, uint32_t row0,
                                              uint32_t k0, uint32_t tensor_rows,
                                              uint32_t lds_byte_off) {
  uint64_t ga = (uint64_t)(uintptr_t)gbase +
                ((uint64_t)row0 * (uint64_t)K_DIM + (uint64_t)k0) * 4ull;
  // D# group 0 (128b): [1:0]=count=1, [63:32]=lds_addr, [120:64]=global_addr,
  // [127:126]=type=2.
  v4u g0;
  g0[0] = 1u;
  g0[1] = lds_byte_off;
  g0[2] = (uint32_t)ga;
  g0[3] = (uint32_t)((ga >> 32) & 0x1FFFFFFull) | (2u << 30);
  // D# group 1 (256b): data_size=2(4B), pad_enable=1, pad_interval=4(32dw),
  // pad_amount=3(4dw); tensor_dim0=K_DIM, tensor_dim1=tensor_rows;
  // tile_dim0=KC, tile_dim1=64, tile_dim2=0; tensor_dim0_stride=K_DIM.
  v8i g1;
  g1[0] = (int)((2u << 16) | (1u << 20) | (4u << 22) | (3u << 25));
  g1[1] = (int)(((uint32_t)K_DIM & 0xFFFFu) << 16);           // dim0 lo16 in [31:16]
  g1[2] = (int)(((uint32_t)K_DIM >> 16) | ((tensor_rows & 0xFFFFu) << 16));
  g1[3] = (int)((tensor_rows >> 16) | ((uint32_t)KC << 16));  // dim1 hi16 | tile_dim0
  g1[4] = (int)WT;                                            // tile_dim1=64, tile_dim2=0
  g1[5] = (int)K_DIM;                                         // dim0_stride lo32
  g1[6] = 0;
  g1[7] = 0;
  v4i z4 = {0, 0, 0, 0};                 // groups 2/3 unused (2D tensor)
  v8i z8 = {0, 0, 0, 0, 0, 0, 0, 0};     // extra group (clang-23 6-arg form)
  __builtin_amdgcn_tensor_load_to_lds(g0, g1, z4, z4, z8, 0);
}

__global__ __launch_bounds__(256)
void l1dist_kernel(const float* __restrict__ x, const float* __restrict__ w,
                   const float* __restrict__ bias, float* __restrict__ out) {
  __shared__ float ldsX[2][XT * LDS_STRIDE];
  __shared__ float ldsW[2][WT * LDS_STRIDE];

  const int tid = threadIdx.x;
  const int tx  = tid & 15;        // w-row group
  const int ty  = tid >> 4;        // x-row group
  const int bx  = blockIdx.x;      // x tile index (N/64 = 32)
  const int by  = blockIdx.y;      // w tile index (C/64 = 64)
  const bool wave0 = (tid < 32);   // wave-uniform predicate

  // Prologue: wave 0 kicks off chunk 0 into buffer 0.
  if (wave0) {
    tdm_load_tile(x, (uint32_t)(bx * XT), 0u, (uint32_t)N_ROWS,
                  (uint32_t)(uintptr_t)&ldsX[0][0]);
    tdm_load_tile(w, (uint32_t)(by * WT), 0u, (uint32_t)C_COLS,
                  (uint32_t)(uintptr_t)&ldsW[0][0]);
  }

  float acc[4][4];
#pragma unroll
  for (int i = 0; i < 4; ++i)
#pragma unroll
    for (int j = 0; j < 4; ++j) acc[i][j] = 0.0f;

  const int xr = ty * 4;
  const int wr = tx * 4;

  for (int ci = 0; ci < NCHUNK; ++ci) {
    const int b = ci & 1;
    if (wave0) {
      if (ci + 1 < NCHUNK) {
        // Prefetch next chunk into the back buffer (safe: all waves finished
        // computing on it at the barrier that ended iteration ci-1).
        tdm_load_tile(x, (uint32_t)(bx * XT), (uint32_t)((ci + 1) * KC),
                      (uint32_t)N_ROWS, (uint32_t)(uintptr_t)&ldsX[1 - b][0]);
        tdm_load_tile(w, (uint32_t)(by * WT), (uint32_t)((ci + 1) * KC),
                      (uint32_t)C_COLS, (uint32_t)(uintptr_t)&ldsW[1 - b][0]);
        // TENSORcnt is in-order per wave: <=2 outstanding means the two loads
        // of the *current* chunk have landed in LDS.
        __builtin_amdgcn_s_wait_tensorcnt(2);
      } else {
        __builtin_amdgcn_s_wait_tensorcnt(0);
      }
    }
    __syncthreads();

    const float* xb = &ldsX[b][0];
    const float* wb = &ldsW[b][0];
#pragma unroll
    for (int k4 = 0; k4 < KC; k4 += 4) {
      float4 a[4], c[4];
#pragma unroll
      for (int i = 0; i < 4; ++i)
        a[i] = *(const float4*)(xb + (xr + i) * LDS_STRIDE + k4);
#pragma unroll
      for (int j = 0; j < 4; ++j)
        c[j] = *(const float4*)(wb + (wr + j) * LDS_STRIDE + k4);
#pragma unroll
      for (int i = 0; i < 4; ++i)
#pragma unroll
        for (int j = 0; j < 4; ++j) {
          acc[i][j] += __builtin_fabsf(a[i].x - c[j].x);
          acc[i][j] += __builtin_fabsf(a[i].y - c[j].y);
          acc[i][j] += __builtin_fabsf(a[i].z - c[j].z);
          acc[i][j] += __builtin_fabsf(a[i].w - c[j].w);
        }
    }
    __syncthreads();
  }

  // Epilogue: out[n,c] = bias[c] - sum|.|, vectorized float4 stores.
  const int n0 = bx * XT + xr;
  const int c0 = by * WT + wr;
  const float4 bv = *(const float4*)(bias + c0);
#pragma unroll
  for (int i = 0; i < 4; ++i) {
    float4 o;
    o.x = bv.x - acc[i][0];
    o.y = bv.y - acc[i][1];
    o.z = bv.z - acc[i][2];
    o.w = bv.w - acc[i][3];
    *(float4*)(out + (size_t)(n0 + i) * C_COLS + c0) = o;
  }
}

extern "C" void kernel_launch(void* const* d_in, const int* in_sizes, int n_in,
                              void* d_out, int out_size, void* d_ws, size_t ws_size,
                              hipStream_t stream) {
  const float* x    = (const float*)d_in[0];   // (2, 1024, 1024) f32
  const float* wpat = (const float*)d_in[1];   // (4096, 1024) f32
  const float* bias = (const float*)d_in[2];   // (4096,) f32
  float* out = (float*)d_out;                  // (2, 1024, 4096) f32

  dim3 grid(N_ROWS / XT, C_COLS / WT);         // 32 x 64 = 2048 blocks
  dim3 block(256);                             // 8 wave32
  l1dist_kernel<<<grid, block, 0, stream>>>(x, wpat, bias, out);
}